// _QuantumFilter_65481071409281
// MI455X (gfx1250) — compile-verified
//
#include <hip/hip_runtime.h>
#include <stdint.h>

// out[n,h,w] = cos(pi * x[n,2h,2w] / (m[h,w]+1e-8)),
// m[h,w] = max_n max(x[n,2h,2w], x[n,2h,2w+1]).
// (Unit-modulus phases + CNOT perm applied twice == identity on |a|^2,
//  so params drops out and probs.Z0 == cos^2 - sin^2 == cos(t0).)

#define N_IMG      8192
#define IN_SLICE   3136      // 56*56 floats per image
#define IN_BYTES   12544     // bytes per image
#define OUTP       784       // 28*28 outputs per image
#define BUF_FLOATS 1568      // one slice's even rows: 392 chunks * 4 floats
#define BUF_BYTES  6272      // 392 chunks * 16B
#define NBUF       4         // async pipeline depth (prefetch 3 ahead)
#define B_BLOCKS   512
#define B_SLICES   16        // 512*16 = 8192
#define C_BLOCKS   1024
#define C_SLICES   8         // 1024*8 = 8192

__device__ __forceinline__ void async_b128(uint32_t lds_off, uint32_t goff, const void* base) {
    // GVS mode: mem = SADDR + VADDR(i32) + inst_offset ; VDST vgpr = LDS byte addr
    asm volatile("global_load_async_to_lds_b128 %0, %1, %2 offset:0"
                 :: "v"(lds_off), "v"(goff), "s"(base) : "memory");
}

__global__ void __launch_bounds__(256) kmax(const float* __restrict__ x,
                                            float* __restrict__ m) {
    __shared__ float buf[NBUF * BUF_FLOATS];     // 4 * 6272 B = 25088 B
    const int t = threadIdx.x;
    const int w = t >> 5;                        // wave id (wave32)
    const int l = t & 31;                        // lane id
    const int n0 = blockIdx.x * B_SLICES;

    // Wave w owns chunks [49w, 49w+49); chunk k covers positions p=2k,2k+1 and
    // sits at in-slice byte (k/14)*448 + (k%14)*16 (even rows only).
    const int k0 = 49 * w + l;                   // always < 392
    const int k1 = k0 + 32;                      // valid iff l < 17
    const uint32_t lds_base = (uint32_t)(uintptr_t)&buf[0];  // low dword = LDS offset
    const uint32_t c0 = lds_base + (uint32_t)k0 * 16u;
    const uint32_t c1 = lds_base + (uint32_t)k1 * 16u;
    const uint32_t g0 = (uint32_t)((k0 / 14) * 448 + (k0 % 14) * 16);
    const uint32_t g1 = (uint32_t)((k1 / 14) * 448 + (k1 % 14) * 16);

    const int p0 = 98 * w + l, p1 = p0 + 32, p2 = p0 + 64, p3 = p0 + 96; // p3: l<2
    float a0 = 0.f, a1 = 0.f, a2 = 0.f, a3 = 0.f;

    auto issue = [&](int n, int bb) {            // 2 async instrs per wave, always
        const uint32_t gb = (uint32_t)n * IN_BYTES;
        const uint32_t lb = (uint32_t)bb * BUF_BYTES;
        async_b128(c0 + lb, gb + g0, x);
        if (l < 17) async_b128(c1 + lb, gb + g1, x);   // exec=0x1FFFF, still issues
    };
    auto consume = [&](int i) {
        const float* bp = &buf[(i & (NBUF - 1)) * BUF_FLOATS];
        float2 v0 = *(const float2*)(bp + 2 * p0);
        float2 v1 = *(const float2*)(bp + 2 * p1);
        float2 v2 = *(const float2*)(bp + 2 * p2);
        a0 = fmaxf(a0, fmaxf(v0.x, v0.y));
        a1 = fmaxf(a1, fmaxf(v1.x, v1.y));
        a2 = fmaxf(a2, fmaxf(v2.x, v2.y));
        if (l < 2) {
            float2 v3 = *(const float2*)(bp + 2 * p3);
            a3 = fmaxf(a3, fmaxf(v3.x, v3.y));
        }
    };

    // Prologue: prefetch 3 slices (asynccnt = 6).
    issue(n0 + 0, 0);
    issue(n0 + 1, 1);
    issue(n0 + 2, 2);

    // Steady state: issue slice i+3, wait until slice i landed (6 younger
    // instrs outstanding; async loads complete in order), consume slice i.
    for (int i = 0; i < B_SLICES - 3; ++i) {
        asm volatile("s_wait_dscnt 0" ::: "memory");      // buffer reuse safety
        issue(n0 + i + 3, (i + 3) & (NBUF - 1));
        asm volatile("s_wait_asynccnt 6" ::: "memory");
        consume(i);
    }
    // Tail: drain pipeline with constant wait immediates.
    asm volatile("s_wait_asynccnt 4" ::: "memory"); consume(B_SLICES - 3);
    asm volatile("s_wait_asynccnt 2" ::: "memory"); consume(B_SLICES - 2);
    asm volatile("s_wait_asynccnt 0" ::: "memory"); consume(B_SLICES - 1);

    // non-negative floats: uint compare == float compare -> global_atomic_max_u32
    atomicMax((unsigned int*)&m[p0], __float_as_uint(a0));
    atomicMax((unsigned int*)&m[p1], __float_as_uint(a1));
    atomicMax((unsigned int*)&m[p2], __float_as_uint(a2));
    if (l < 2) atomicMax((unsigned int*)&m[p3], __float_as_uint(a3));
}

__global__ void __launch_bounds__(256) kout(const float* __restrict__ x,
                                            const float* __restrict__ m,
                                            float* __restrict__ out) {
    const int t = threadIdx.x;
    const int n0 = blockIdx.x * C_SLICES;
    const int p0 = t, p1 = t + 256, p2 = t + 512, p3 = t + 768;  // p3: t<16

    // v_cos_f32 has period 1.0, so fold the *0.5 into the reciprocal:
    // out = cos(2*pi * (x * 0.5/(m+1e-8)))
    const float s0 = 0.5f / (m[p0] + 1e-8f);
    const float s1 = 0.5f / (m[p1] + 1e-8f);
    const float s2 = 0.5f / (m[p2] + 1e-8f);
    const float s3 = (t < 16) ? 0.5f / (m[p3] + 1e-8f) : 0.f;
    const int o0 = (p0 / 28) * 112 + (p0 % 28) * 2;
    const int o1 = (p1 / 28) * 112 + (p1 % 28) * 2;
    const int o2 = (p2 / 28) * 112 + (p2 % 28) * 2;
    const int o3 = (t < 16) ? (p3 / 28) * 112 + (p3 % 28) * 2 : 0;

#pragma unroll
    for (int s = 0; s < C_SLICES; ++s) {
        const float* xs = x + (size_t)(n0 + s) * IN_SLICE;
        float* os = out + (size_t)(n0 + s) * OUTP;
        float2 v;
        v = *(const float2*)(xs + o0); os[p0] = __builtin_amdgcn_cosf(v.x * s0);
        v = *(const float2*)(xs + o1); os[p1] = __builtin_amdgcn_cosf(v.x * s1);
        v = *(const float2*)(xs + o2); os[p2] = __builtin_amdgcn_cosf(v.x * s2);
        if (t < 16) {
            v = *(const float2*)(xs + o3); os[p3] = __builtin_amdgcn_cosf(v.x * s3);
        }
    }
}

extern "C" void kernel_launch(void* const* d_in, const int* in_sizes, int n_in,
                              void* d_out, int out_size, void* d_ws, size_t ws_size,
                              hipStream_t stream) {
    const float* x = (const float*)d_in[0];
    // d_in[1] (params) is provably unused: unit-modulus phases + involutive
    // permutation applied twice leave |amplitudes|^2 unchanged.
    (void)in_sizes; (void)n_in; (void)out_size; (void)ws_size;

    float* mbuf = (float*)d_ws;                 // 784 floats of scratch
    float* out  = (float*)d_out;

    hipMemsetAsync(d_ws, 0, OUTP * sizeof(float), stream);   // identity for max (x >= 0)
    kmax<<<B_BLOCKS, 256, 0, stream>>>(x, mbuf);
    kout<<<C_BLOCKS, 256, 0, stream>>>(x, mbuf, out);
}